// DGLGATSubgraphEncoder_19250043421249
// MI455X (gfx1250) — compile-verified
//
#include <hip/hip_runtime.h>
#include <math.h>

#define NH_HEADS 4
#define DFEAT    64
#define HD       256
#define OUTF     128
#define NEG_SLOPE 0.2f

typedef __attribute__((ext_vector_type(2))) float v2f;
typedef __attribute__((ext_vector_type(8))) float v8f;

// ---- order-preserving float<->uint encoding for atomicMax on f32 ----
__device__ __forceinline__ unsigned fenc(float f) {
    unsigned u = __float_as_uint(f);
    return (u & 0x80000000u) ? ~u : (u | 0x80000000u);
}
__device__ __forceinline__ float fdec(unsigned k) {
    return (k & 0x80000000u) ? __uint_as_float(k ^ 0x80000000u)
                             : __uint_as_float(~k);
}
#define ENC_NEG_INF 0x007FFFFFu   // fenc(-inf)

// ---------------------------------------------------------------------
// ws layout (floats): [0, NH)      emax (as unsigned, encoded)
//                     [NH, 2NH)    denom
//                     [2NH, 3NH)   num
//                     [3NH, 3NH+4) sacc[h]   (Σ_n s[n,h])
//                     [3NH+4, +8)  cl[h]
//                     [3NH+8, +12) cr[h]
// ---------------------------------------------------------------------

__global__ void gat_init_kernel(unsigned* emax, float* denom, float* num,
                                float* consts, int NH) {
    int i = blockIdx.x * blockDim.x + threadIdx.x;
    if (i < NH) {
        emax[i]  = ENC_NEG_INF;
        denom[i] = 0.f;
        num[i]   = 0.f;
    }
    if (i < 12) consts[i] = 0.f;
}

// cl[h] = Σ_d W[h*64+d]*attn_l[h*64+d], cr likewise. One 256-thread block.
__global__ void gat_attn_const_kernel(const float* __restrict__ W,
                                      const float* __restrict__ attn_l,
                                      const float* __restrict__ attn_r,
                                      float* consts) {
    __shared__ float pl[HD], pr[HD];
    int t = threadIdx.x;
    float w = W[t];
    pl[t] = w * attn_l[t];
    pr[t] = w * attn_r[t];
    __syncthreads();
    if (t < NH_HEADS) {
        float sl = 0.f, sr = 0.f;
        for (int d = 0; d < DFEAT; ++d) {
            sl += pl[t * DFEAT + d];
            sr += pr[t * DFEAT + d];
        }
        consts[4 + t] = sl;
        consts[8 + t] = sr;
    }
}

__device__ __forceinline__ float edge_score(float fs, float fd, float cl, float cr) {
    float x = fs * cl + fd * cr;
    return (x >= 0.f) ? x : NEG_SLOPE * x;
}

// Pass 1: segment max of leaky-relu scores into emax[dst,h]
__global__ void gat_edge_max_kernel(const int* __restrict__ src,
                                    const int* __restrict__ dst,
                                    const float* __restrict__ feat,
                                    const float* __restrict__ consts,
                                    unsigned* emax, int E) {
    int e = blockIdx.x * blockDim.x + threadIdx.x;
    if (e >= E) return;
    int s = src[e], d = dst[e];
    float fs = feat[s], fd = feat[d];
#pragma unroll
    for (int h = 0; h < NH_HEADS; ++h) {
        float x = edge_score(fs, fd, consts[4 + h], consts[8 + h]);
        atomicMax(&emax[d * NH_HEADS + h], fenc(x));
    }
}

// Pass 2: ee = exp(score - max); denom += ee; num += ee * f[src]
__global__ void gat_edge_sum_kernel(const int* __restrict__ src,
                                    const int* __restrict__ dst,
                                    const float* __restrict__ feat,
                                    const float* __restrict__ consts,
                                    const unsigned* __restrict__ emax,
                                    float* denom, float* num, int E) {
    int e = blockIdx.x * blockDim.x + threadIdx.x;
    if (e >= E) return;
    int s = src[e], d = dst[e];
    float fs = feat[s], fd = feat[d];
#pragma unroll
    for (int h = 0; h < NH_HEADS; ++h) {
        float x  = edge_score(fs, fd, consts[4 + h], consts[8 + h]);
        float mx = fdec(emax[d * NH_HEADS + h]);
        float ee = __expf(x - mx);
        atomicAdd(&denom[d * NH_HEADS + h], ee);
        atomicAdd(&num[d * NH_HEADS + h], ee * fs);
    }
}

// Per-node s[n,h] = num/denom (0 if no in-edges), block-reduce, accumulate sacc[h]
__global__ void gat_node_reduce_kernel(const float* __restrict__ denom,
                                       const float* __restrict__ num,
                                       float* consts, int N) {
    __shared__ float red[256];
    int n = blockIdx.x * blockDim.x + threadIdx.x;
    int t = threadIdx.x;
#pragma unroll
    for (int h = 0; h < NH_HEADS; ++h) {
        float v = 0.f;
        if (n < N) {
            float dn = denom[n * NH_HEADS + h];
            if (dn > 0.f) v = num[n * NH_HEADS + h] / dn;
        }
        red[t] = v;
        __syncthreads();
        for (int stride = 128; stride > 0; stride >>= 1) {
            if (t < stride) red[t] += red[t + stride];
            __syncthreads();
        }
        if (t == 0) atomicAdd(&consts[h], red[0]);
        __syncthreads();
    }
}

// Final: m[k] = sbar[h(k)]*W[k] + bias_gat[k]; out = m @ fc_W + fc_b via
// V_WMMA_F32_16X16X4_F32. 8 waves, one 16-col tile each; A holds m in row M=0.
__global__ void gat_final_wmma_kernel(const float* __restrict__ W,
                                      const float* __restrict__ bias_gat,
                                      const float* __restrict__ fcW,
                                      const float* __restrict__ fcb,
                                      const float* __restrict__ consts,
                                      float* __restrict__ out, float invN) {
    __shared__ float m[HD];
    int t = threadIdx.x;
    int h = t / DFEAT;
    m[t] = consts[h] * invN * W[t] + bias_gat[t];
    __syncthreads();

    int wave = t >> 5;        // 0..7  -> N-tile
    int lane = t & 31;
    int j0   = wave * 16;

    v8f c = {};
#pragma unroll 4
    for (int kk = 0; kk < HD / 4; ++kk) {
        int k0 = kk * 4;
        // A fragment (16x4 f32): lanes 0-15 carry K=0,1; lanes 16-31 carry K=2,3.
        // Only row M=0 is non-zero (the mean hidden vector).
        v2f a;
        a.x = 0.f; a.y = 0.f;
        if (lane == 0)  { a.x = m[k0 + 0]; a.y = m[k0 + 1]; }
        if (lane == 16) { a.x = m[k0 + 2]; a.y = m[k0 + 3]; }
        // B fragment (4x16 f32): mirror layout, rows striped across lanes.
        int col  = lane & 15;
        int krow = (lane < 16) ? 0 : 2;
        v2f b;
        b.x = fcW[(k0 + krow)     * OUTF + j0 + col];
        b.y = fcW[(k0 + krow + 1) * OUTF + j0 + col];
        c = __builtin_amdgcn_wmma_f32_16x16x4_f32(
                /*neg_a=*/false, a, /*neg_b=*/false, b,
                /*c_mod=*/(short)0, c, /*reuse_a=*/false, /*reuse_b=*/false);
    }
    // D row M=0 lives in VGPR0 lanes 0-15.
    if (lane < 16) out[j0 + lane] = c[0] + fcb[j0 + lane];
}

extern "C" void kernel_launch(void* const* d_in, const int* in_sizes, int n_in,
                              void* d_out, int out_size, void* d_ws, size_t ws_size,
                              hipStream_t stream) {
    const float* features = (const float*)d_in[0];
    const float* W        = (const float*)d_in[1];
    const float* attn_l   = (const float*)d_in[2];
    const float* attn_r   = (const float*)d_in[3];
    const float* bias_gat = (const float*)d_in[4];
    const float* fc_W     = (const float*)d_in[5];
    const float* fc_b     = (const float*)d_in[6];
    const int*   src      = (const int*)d_in[7];
    const int*   dst      = (const int*)d_in[8];

    const int N  = in_sizes[0];     // features is [N,1]
    const int E  = in_sizes[7];
    const int NH = N * NH_HEADS;

    unsigned* emax  = (unsigned*)d_ws;
    float*    denom = (float*)d_ws + NH;
    float*    num   = (float*)d_ws + 2 * NH;
    float*    cst   = (float*)d_ws + 3 * NH;   // sacc[4], cl[4], cr[4]

    const int TB = 256;
    gat_init_kernel<<<(NH + TB - 1) / TB, TB, 0, stream>>>(emax, denom, num, cst, NH);
    gat_attn_const_kernel<<<1, HD, 0, stream>>>(W, attn_l, attn_r, cst);
    gat_edge_max_kernel<<<(E + TB - 1) / TB, TB, 0, stream>>>(src, dst, features, cst, emax, E);
    gat_edge_sum_kernel<<<(E + TB - 1) / TB, TB, 0, stream>>>(src, dst, features, cst, emax,
                                                              denom, num, E);
    gat_node_reduce_kernel<<<(N + TB - 1) / TB, TB, 0, stream>>>(denom, num, cst, N);
    gat_final_wmma_kernel<<<1, 256, 0, stream>>>(W, bias_gat, fc_W, fc_b, cst,
                                                 (float*)d_out, 1.0f / (float)N);
}